// GAT_88252987998923
// MI455X (gfx1250) — compile-verified
//
#include <hip/hip_runtime.h>
#include <hip/hip_bf16.h>

#define N_NODES 100000
#define D_FEAT  128
#define N_EDGES 600000
#define N_SETS  2
#define HID_SZ  80

typedef float v2f __attribute__((ext_vector_type(2)));
typedef float v8f __attribute__((ext_vector_type(8)));

// ---------------- order-preserving float<->uint encoding for atomic max ----
__device__ __forceinline__ unsigned fenc(float f) {
    int i = __float_as_int(f);
    return (i >= 0) ? ((unsigned)i | 0x80000000u) : ~((unsigned)i);
}
__device__ __forceinline__ float fdec(unsigned e) {
    int i = (e & 0x80000000u) ? (int)(e & 0x7FFFFFFFu) : (int)(~e);
    return __int_as_float(i);
}

// ---------------- u = W@a1, v = W@a2 for both s (512 dots of len 128) ------
__global__ void gat_uv_kernel(const float* __restrict__ W,
                              const float* __restrict__ att_w,
                              float* __restrict__ uv) {
    int t = blockIdx.x * blockDim.x + threadIdx.x;
    if (t >= 2 * 2 * D_FEAT) return;
    int s     = t >> 8;          // set index
    int which = (t >> 7) & 1;    // 0 -> u (a1 half), 1 -> v (a2 half)
    int i     = t & 127;
    const float* a  = att_w + which * D_FEAT;
    const float* Wr = W + (size_t)s * D_FEAT * D_FEAT + (size_t)i * D_FEAT;
    float acc = 0.f;
    for (int j = 0; j < D_FEAT; ++j) acc += Wr[j] * a[j];
    uv[t] = acc;   // layout: [s][which][128]
}

// ---------------- y = h @ W[s] via V_WMMA_F32_16X16X4_F32 ------------------
// Per wave: one 16-row strip, 8 column tiles (16 cols each), K loop step 4.
__global__ void __launch_bounds__(256)
gat_gemm_kernel(const float* __restrict__ h,
                const float* __restrict__ W,   // this set's 128x128
                float* __restrict__ y,
                int nStrips) {
    __shared__ float Wlds[D_FEAT * D_FEAT];   // 64 KB
    const int tid = threadIdx.x;
    {   // cooperative load of W into LDS (float4)
        const float4* Wg = (const float4*)W;
        float4*       Wl = (float4*)Wlds;
        for (int i = tid; i < (D_FEAT * D_FEAT) / 4; i += 256) Wl[i] = Wg[i];
    }
    __syncthreads();

    const int wave  = tid >> 5;
    const int lane  = tid & 31;
    const int strip = blockIdx.x * 8 + wave;
    if (strip >= nStrips) return;             // wave-uniform: EXEC stays all-1

    const int r0    = strip * 16;
    const int lrow  = lane & 15;
    const int khalf = (lane >> 4) << 1;       // 0 or 2 (K sub-offset)

    v8f acc[8] = {};
    const float* hrow = h + (size_t)(r0 + lrow) * D_FEAT + khalf;

    for (int k0 = 0; k0 < D_FEAT; k0 += 4) {
        v2f a;                                 // A 16x4: lanes0-15 K={k0,k0+1}, lanes16-31 K={k0+2,k0+3}
        a.x = hrow[k0];
        a.y = hrow[k0 + 1];
#pragma unroll
        for (int t = 0; t < 8; ++t) {
            const int col = t * 16 + lrow;
            v2f b;                             // B 4x16, same K split across lane halves
            b.x = Wlds[(k0 + khalf) * D_FEAT + col];
            b.y = Wlds[(k0 + khalf + 1) * D_FEAT + col];
            acc[t] = __builtin_amdgcn_wmma_f32_16x16x4_f32(
                false, a, false, b, (short)0, acc[t], false, false);
        }
    }
    // D 16x16: VGPR v -> row v (lanes0-15) / row v+8 (lanes16-31), col = lane&15
    const int rhalf = (lane >> 4) << 3;
#pragma unroll
    for (int t = 0; t < 8; ++t)
#pragma unroll
        for (int v = 0; v < 8; ++v)
            y[(size_t)(r0 + rhalf + v) * D_FEAT + t * 16 + lrow] = acc[t][v];
}

// ---------------- p[n]=h[n]·u, q[n]=h[n]·v (wave per node) -----------------
__global__ void gat_pq_kernel(const float* __restrict__ h,
                              const float* __restrict__ uv,  // u(128), v(128)
                              float* __restrict__ p,
                              float* __restrict__ q,
                              int n) {
    const int gw   = (blockIdx.x * blockDim.x + threadIdx.x) >> 5;
    const int lane = threadIdx.x & 31;
    if (gw >= n) return;
    const float4 h4 = ((const float4*)(h + (size_t)gw * D_FEAT))[lane];
    const float4 u4 = ((const float4*)uv)[lane];
    const float4 v4 = ((const float4*)(uv + D_FEAT))[lane];
    float pa = h4.x * u4.x + h4.y * u4.y + h4.z * u4.z + h4.w * u4.w;
    float qa = h4.x * v4.x + h4.y * v4.y + h4.z * v4.z + h4.w * v4.w;
    for (int off = 16; off; off >>= 1) {
        pa += __shfl_down(pa, off, 32);
        qa += __shfl_down(qa, off, 32);
    }
    if (lane == 0) { p[gw] = pa; q[gw] = qa; }
}

// ---------------- pass A: e = lrelu(p[src]+q[dst]+b); segment max ----------
__global__ void gat_logits_kernel(const int* __restrict__ edges,     // [E][2]
                                  const float* __restrict__ p,
                                  const float* __restrict__ q,
                                  const float* __restrict__ att_b,
                                  float* __restrict__ elog,
                                  unsigned* __restrict__ mEnc,
                                  int nE) {
    int e = blockIdx.x * blockDim.x + threadIdx.x;
    if (e >= nE) return;
    int src = edges[2 * e], dst = edges[2 * e + 1];
    float l = p[src] + q[dst] + att_b[0];
    l = (l > 0.f) ? l : 0.01f * l;            // leaky_relu, slope 0.01
    elog[e] = l;
    atomicMax(&mEnc[dst], fenc(l));
}

// ---------------- pass B: ex = exp(e - m[dst]); segment sum ----------------
__global__ void gat_exp_kernel(const int* __restrict__ edges,
                               const unsigned* __restrict__ mEnc,
                               float* __restrict__ elog,
                               float* __restrict__ sums,
                               int nE) {
    int e = blockIdx.x * blockDim.x + threadIdx.x;
    if (e >= nE) return;
    int dst = edges[2 * e + 1];
    float ex = expf(elog[e] - fdec(mEnc[dst]));
    elog[e] = ex;
    atomicAdd(&sums[dst], ex);
}

// ---------------- pass C: h_prime[src] += alpha * y[dst] (32 lanes/edge) ---
__global__ void gat_scatter_kernel(const int* __restrict__ edges,
                                   const float* __restrict__ elog,
                                   const float* __restrict__ sums,
                                   const float* __restrict__ y,
                                   float* __restrict__ h_prime,
                                   int nE) {
    long long tid = (long long)blockIdx.x * blockDim.x + threadIdx.x;
    int e    = (int)(tid >> 5);
    int lane = (int)(tid & 31);
    if (e >= nE) return;
    int src = edges[2 * e], dst = edges[2 * e + 1];
    float alpha = elog[e] / sums[dst];
    const float4 yv = ((const float4*)(y + (size_t)dst * D_FEAT))[lane];
    float* hp = h_prime + (size_t)src * D_FEAT + lane * 4;
    atomicAdd(hp + 0, alpha * yv.x);
    atomicAdd(hp + 1, alpha * yv.y);
    atomicAdd(hp + 2, alpha * yv.z);
    atomicAdd(hp + 3, alpha * yv.w);
}

// ---------------- column sums of h_prime -> gsum ---------------------------
__global__ void gat_mean_kernel(const float* __restrict__ h_prime,
                                float* __restrict__ gsum, int n) {
    int col = threadIdx.x & 127;
    int rib = threadIdx.x >> 7;                 // 0..1 (256 threads)
    float acc = 0.f;
    for (int r = blockIdx.x * 2 + rib; r < n; r += gridDim.x * 2)
        acc += h_prime[(size_t)r * D_FEAT + col];
    atomicAdd(&gsum[col], acc);
}

// ---------------- tiny MLP head (single block) -----------------------------
__global__ void gat_head_kernel(const float* __restrict__ gsum,
                                const float* __restrict__ ptype,
                                const float* __restrict__ fc1w, const float* __restrict__ fc1b,
                                const float* __restrict__ fc2w, const float* __restrict__ fc2b,
                                const float* __restrict__ fc3w, const float* __restrict__ fc3b,
                                float* __restrict__ out, float invN) {
    __shared__ float x[D_FEAT + 1];
    __shared__ float h1[HID_SZ];
    __shared__ float h2[HID_SZ];
    int t = threadIdx.x;
    if (t < D_FEAT) x[t] = gsum[t] * invN;
    if (t == D_FEAT) x[D_FEAT] = ptype[0];
    __syncthreads();
    if (t < HID_SZ) {
        float acc = fc1b[t];
        for (int j = 0; j < D_FEAT + 1; ++j) acc += fc1w[t * (D_FEAT + 1) + j] * x[j];
        h1[t] = (acc > 0.f) ? acc : 0.01f * acc;
    }
    __syncthreads();
    if (t < HID_SZ) {
        float acc = fc2b[t];
        for (int j = 0; j < HID_SZ; ++j) acc += fc2w[t * HID_SZ + j] * h1[j];
        h2[t] = (acc > 0.f) ? acc : 0.01f * acc;
    }
    __syncthreads();
    if (t < 2) {
        float acc = fc3b[t];
        for (int j = 0; j < HID_SZ; ++j) acc += fc3w[t * HID_SZ + j] * h2[j];
        out[t] = acc;
    }
}

extern "C" void kernel_launch(void* const* d_in, const int* in_sizes, int n_in,
                              void* d_out, int out_size, void* d_ws, size_t ws_size,
                              hipStream_t stream) {
    const float* nodes  = (const float*)d_in[0];
    const int*   edges  = (const int*)d_in[1];
    const float* ptype  = (const float*)d_in[2];
    const float* W      = (const float*)d_in[3];
    const float* att_w  = (const float*)d_in[4];
    const float* att_b  = (const float*)d_in[5];
    const float* fc1w   = (const float*)d_in[6];
    const float* fc1b   = (const float*)d_in[7];
    const float* fc2w   = (const float*)d_in[8];
    const float* fc2b   = (const float*)d_in[9];
    const float* fc3w   = (const float*)d_in[10];
    const float* fc3b   = (const float*)d_in[11];
    float* out = (float*)d_out;

    // ---- workspace layout (floats) ----
    float* ws = (float*)d_ws;
    float*    y       = ws;                               // N*D
    float*    h_prime = y + (size_t)N_NODES * D_FEAT;     // N*D
    float*    p       = h_prime + (size_t)N_NODES * D_FEAT; // N
    float*    q       = p + N_NODES;                      // N
    float*    elog    = q + N_NODES;                      // E
    unsigned* mEnc    = (unsigned*)(elog + N_EDGES);      // S*N
    float*    sums    = (float*)(mEnc + N_SETS * N_NODES); // S*N
    float*    uv      = sums + N_SETS * N_NODES;          // 2*2*128
    float*    gsum    = uv + 512;                         // 128

    // ---- zero accumulators (graph-capture-safe memsets) ----
    hipMemsetAsync(h_prime, 0, (size_t)N_NODES * D_FEAT * sizeof(float), stream);
    // mEnc: 0 is below every encoded float (encodings >= 0x007FFFFF) -> acts as -inf
    hipMemsetAsync(mEnc, 0, (size_t)N_SETS * N_NODES * sizeof(unsigned), stream);
    hipMemsetAsync(sums, 0, (size_t)N_SETS * N_NODES * sizeof(float), stream);
    hipMemsetAsync(gsum, 0, D_FEAT * sizeof(float), stream);

    // ---- u,v vectors ----
    gat_uv_kernel<<<2, 256, 0, stream>>>(W, att_w, uv);

    const int nStrips = N_NODES / 16;                     // 6250 exactly
    for (int s = 0; s < N_SETS; ++s) {
        const float* Ws = W + (size_t)s * D_FEAT * D_FEAT;
        const int*   Es = edges + (size_t)s * N_EDGES * 2;
        const float* uvS = uv + (size_t)s * 2 * D_FEAT;
        unsigned* mS = mEnc + (size_t)s * N_NODES;
        float*    sS = sums + (size_t)s * N_NODES;

        gat_gemm_kernel<<<(nStrips + 7) / 8, 256, 0, stream>>>(nodes, Ws, y, nStrips);
        gat_pq_kernel<<<(N_NODES + 7) / 8, 256, 0, stream>>>(nodes, uvS, p, q, N_NODES);
        gat_logits_kernel<<<(N_EDGES + 255) / 256, 256, 0, stream>>>(Es, p, q, att_b, elog, mS, N_EDGES);
        gat_exp_kernel<<<(N_EDGES + 255) / 256, 256, 0, stream>>>(Es, mS, elog, sS, N_EDGES);
        {
            long long threads = (long long)N_EDGES * 32;
            gat_scatter_kernel<<<(int)((threads + 255) / 256), 256, 0, stream>>>(
                Es, elog, sS, y, h_prime, N_EDGES);
        }
    }

    gat_mean_kernel<<<512, 256, 0, stream>>>(h_prime, gsum, N_NODES);
    gat_head_kernel<<<1, 256, 0, stream>>>(gsum, ptype, fc1w, fc1b, fc2w, fc2b,
                                           fc3w, fc3b, out, 1.0f / (float)N_NODES);
}